// TransE_38903813767663
// MI455X (gfx1250) — compile-verified
//
#include <hip/hip_runtime.h>
#include <math.h>

typedef __attribute__((ext_vector_type(2))) float v2f;
typedef __attribute__((ext_vector_type(8))) float v8f;

#define DIM 64

// ---------------------------------------------------------------------------
// Kernel 1: emb_hr[row] = ent_emb[h[row]] + rel_emb[r[row]], plus ||emb_hr||^2.
// One wave32 per row; each lane handles 2 consecutive floats (64 = 32*2).
// ---------------------------------------------------------------------------
__global__ void hr_kernel(const float* __restrict__ ent,
                          const float* __restrict__ rel,
                          const int* __restrict__ h,
                          const int* __restrict__ r,
                          float* __restrict__ hr,
                          float* __restrict__ hrn,
                          int T) {
  int wave = (int)((blockIdx.x * blockDim.x + threadIdx.x) >> 5);
  int lane = threadIdx.x & 31;
  if (wave >= T) return;
  const float* e = ent + (size_t)h[wave] * DIM;
  const float* q = rel + (size_t)r[wave] * DIM;
  v2f ev = *(const v2f*)(e + 2 * lane);
  v2f qv = *(const v2f*)(q + 2 * lane);
  v2f s;
  s.x = ev.x + qv.x;
  s.y = ev.y + qv.y;
  *(v2f*)(hr + (size_t)wave * DIM + 2 * lane) = s;
  float n = s.x * s.x + s.y * s.y;
  // wave32 butterfly reduction
  for (int off = 16; off > 0; off >>= 1) n += __shfl_xor(n, off, 32);
  if (lane == 0) hrn[wave] = n;
}

// ---------------------------------------------------------------------------
// Kernel 2: entn[row] = ||ent_emb[row]||^2.  One wave32 per entity row.
// ---------------------------------------------------------------------------
__global__ void entnorm_kernel(const float* __restrict__ ent,
                               float* __restrict__ entn,
                               int ENT) {
  int wave = (int)((blockIdx.x * blockDim.x + threadIdx.x) >> 5);
  int lane = threadIdx.x & 31;
  if (wave >= ENT) return;
  const float* e = ent + (size_t)wave * DIM;
  v2f ev = *(const v2f*)(e + 2 * lane);
  float n = ev.x * ev.x + ev.y * ev.y;
  for (int off = 16; off > 0; off >>= 1) n += __shfl_xor(n, off, 32);
  if (lane == 0) entn[wave] = n;
}

// ---------------------------------------------------------------------------
// Kernel 3: GEMM (dot products) via V_WMMA_F32_16X16X4_F32 + fused sqrt-norm
// epilogue. One wave per 16x16 output tile. K = 64 -> 16 WMMA ops per tile.
//
// f32 WMMA operand layout (ISA 7.12.2, 32-bit A 16x4):
//   lane L in [0,16):  VGPR0 = M=L,   K=k+0 ; VGPR1 = M=L,   K=k+1
//   lane L in [16,32): VGPR0 = M=L-16,K=k+2 ; VGPR1 = M=L-16,K=k+3
// B (4x16) mirrors this with N in place of M, so both operands are a plain
// float2 load at row (lane&15), column k + 2*(lane>>4).
//
// C/D layout: VGPR v, lane L -> row m0 + v + 8*(L>>4), col n0 + (L&15).
// ---------------------------------------------------------------------------
__global__ void score_kernel(const float* __restrict__ hr,
                             const float* __restrict__ hrn,
                             const float* __restrict__ ent,
                             const float* __restrict__ entn,
                             float* __restrict__ out,
                             int ENT, int mTiles, int totalWaves) {
  int w = (int)((blockIdx.x * blockDim.x + threadIdx.x) >> 5);
  if (w >= totalWaves) return;  // uniform per wave: EXEC stays all-1s for WMMA
  int lane = threadIdx.x & 31;
  int half = lane >> 4;   // 0 or 1
  int lm   = lane & 15;

  // 4 waves of a block share nTile (same ent_emb tile) for cache reuse.
  int m0 = (w % mTiles) * 16;
  int n0 = (w / mTiles) * 16;

  const float* arow = hr  + (size_t)(m0 + lm) * DIM + 2 * half;
  const float* brow = ent + (size_t)(n0 + lm) * DIM + 2 * half;

  v8f acc = {};
#pragma unroll
  for (int k = 0; k < DIM; k += 4) {
    v2f a = *(const v2f*)(arow + k);
    v2f b = *(const v2f*)(brow + k);
    // 8 args: (neg_a, A, neg_b, B, c_mod, C, reuse_a, reuse_b)
    acc = __builtin_amdgcn_wmma_f32_16x16x4_f32(
        false, a, false, b, (short)0, acc, false, false);
  }

  float bn = entn[n0 + lm];
  int col = n0 + lm;
#pragma unroll
  for (int v = 0; v < 8; ++v) {
    int m = m0 + v + 8 * half;
    float s = hrn[m] + bn - 2.0f * acc[v];
    out[(size_t)m * ENT + col] = sqrtf(fmaxf(s, 0.0f));
  }
}

// ---------------------------------------------------------------------------
// Host launcher. Inputs: ent_emb, rel_emb, h, r, t(unused), batch_size(unused)
// Output: 256*20000 f32 (pos rows 0..127 and neg rows 128..255 concatenated ==
// the full score matrix flattened row-major).
// ---------------------------------------------------------------------------
extern "C" void kernel_launch(void* const* d_in, const int* in_sizes, int n_in,
                              void* d_out, int out_size, void* d_ws, size_t ws_size,
                              hipStream_t stream) {
  const float* ent = (const float*)d_in[0];
  const float* rel = (const float*)d_in[1];
  const int*   h   = (const int*)d_in[2];
  const int*   r   = (const int*)d_in[3];

  const int ENT = in_sizes[0] / DIM;  // 20000
  const int T   = in_sizes[2];        // 256

  float* ws   = (float*)d_ws;
  float* hr   = ws;                        // T*DIM floats
  float* hrn  = ws + (size_t)T * DIM;      // T floats
  float* entn = hrn + T;                   // ENT floats

  // Kernel 1: T waves, 8 waves (256 threads) per block.
  hr_kernel<<<(T + 7) / 8, 256, 0, stream>>>(ent, rel, h, r, hr, hrn, T);

  // Kernel 2: ENT waves.
  entnorm_kernel<<<(ENT + 7) / 8, 256, 0, stream>>>(ent, entn, ENT);

  // Kernel 3: one wave per 16x16 tile; 4 waves (128 threads) per block.
  const int mTiles = T / 16;          // 16
  const int nTiles = ENT / 16;        // 1250
  const int totalWaves = mTiles * nTiles;
  score_kernel<<<(totalWaves + 3) / 4, 128, 0, stream>>>(
      hr, hrn, ent, entn, (float*)d_out, ENT, mTiles, totalWaves);
}